// AmpLoss_87754771792112
// MI455X (gfx1250) — compile-verified
//
#include <hip/hip_runtime.h>
#include <math.h>

// ---------------------------------------------------------------------------
// AmpLoss on MI455X (gfx1250, wave32)
//
// Phase 1: bandwidth-bound streaming pass (268 MB live data -> ~11.5 us floor
//          at 23.3 TB/s). float4 (global_load_b128) coalesced loads, 10 f32
//          accumulators per thread (5 mask counts + 5 masked smape sums),
//          wave32 shuffle tree, LDS cross-wave tree, per-block partials
//          written to d_ws (own slot per block => no atomics, no zero-init,
//          deterministic under graph replay).
// Phase 2: single wave32. Per-lane strided pre-accumulation of block
//          partials, then one V_WMMA_F32_16X16X4_F32 with A = ones performs
//          the final cross-lane column-sum in full f32 precision
//          (D[m][n] = sum_k B[k][n]). Lane 0 evaluates the sequential
//          5-mask recurrence (w = sqrt(log(B/cnt)), carried m_rst).
// ---------------------------------------------------------------------------

typedef float v2f __attribute__((ext_vector_type(2)));
typedef float v8f __attribute__((ext_vector_type(8)));

#define TPB   256   // 8 wave32s per block
#define NBLK  1024  // phase-1 blocks; partials = NBLK * 16 floats = 64 KB ws

__device__ __forceinline__ float smape2(float p, float t) {
    // 2*|p-t| / (|p|+|t|)
    return 2.0f * fabsf(p - t) / (fabsf(p) + fabsf(t));
}

__device__ __forceinline__ void classify_accum(float pe, float ts, float td,
                                               float* cnt, float* sum) {
    bool normal   = (ts < 120.0f) && (td < 80.0f);
    bool elevated = (ts >= 120.0f) && (ts < 130.0f) && (td < 80.0f) && !normal;
    bool h1 = (((ts >= 130.0f) && (ts < 140.0f)) || ((td >= 80.0f) && (td < 90.0f)))
              && !(normal || elevated);
    bool h2 = ((ts >= 140.0f) || (td >= 90.0f)) && !(normal || elevated || h1);
    bool crisis = (ts > 180.0f) || (td > 120.0f);   // intentionally overlapping
    bool mk[5] = {normal, elevated, h1, h2, crisis};
#pragma unroll
    for (int a = 0; a < 5; ++a) {
        cnt[a] += mk[a] ? 1.0f : 0.0f;
        sum[a] += mk[a] ? pe   : 0.0f;
    }
}

__global__ void __launch_bounds__(TPB)
amp_partial(const float* __restrict__ dbp, const float* __restrict__ sbp,
            const float* __restrict__ dt,  const float* __restrict__ st,
            float* __restrict__ part, int nvec /* = B/4 */) {
    const float4* dbp4 = (const float4*)dbp;
    const float4* sbp4 = (const float4*)sbp;
    const float4* dt4  = (const float4*)dt;
    const float4* st4  = (const float4*)st;

    float cnt[5] = {0.f, 0.f, 0.f, 0.f, 0.f};
    float sum[5] = {0.f, 0.f, 0.f, 0.f, 0.f};

    const int tid    = blockIdx.x * TPB + threadIdx.x;
    const int stride = gridDim.x * TPB;

    for (int i = tid; i < nvec; i += stride) {
        float4 pd = dbp4[i];
        float4 ps = sbp4[i];
        float4 td = dt4[i];
        float4 ts = st4[i];

        classify_accum(smape2(pd.x, td.x) + smape2(ps.x, ts.x), ts.x, td.x, cnt, sum);
        classify_accum(smape2(pd.y, td.y) + smape2(ps.y, ts.y), ts.y, td.y, cnt, sum);
        classify_accum(smape2(pd.z, td.z) + smape2(ps.z, ts.z), ts.z, td.z, cnt, sum);
        classify_accum(smape2(pd.w, td.w) + smape2(ps.w, ts.w), ts.w, td.w, cnt, sum);
    }

    // pack 10 accumulators: v[0..4]=counts, v[5..9]=sums
    float v[10];
#pragma unroll
    for (int a = 0; a < 5; ++a) { v[a] = cnt[a]; v[5 + a] = sum[a]; }

    // wave32 shuffle tree
#pragma unroll
    for (int off = 16; off > 0; off >>= 1) {
#pragma unroll
        for (int a = 0; a < 10; ++a)
            v[a] += __shfl_down(v[a], off, 32);
    }

    __shared__ float red[TPB / 32][10];
    const int wave = threadIdx.x >> 5;
    const int lane = threadIdx.x & 31;
    if (lane == 0) {
#pragma unroll
        for (int a = 0; a < 10; ++a) red[wave][a] = v[a];
    }
    __syncthreads();

    if (threadIdx.x == 0) {
        float t[10];
#pragma unroll
        for (int a = 0; a < 10; ++a) {
            float acc = 0.0f;
#pragma unroll
            for (int w = 0; w < TPB / 32; ++w) acc += red[w][a];
            t[a] = acc;
        }
        float* out = part + (size_t)blockIdx.x * 16;
#pragma unroll
        for (int a = 0; a < 10; ++a) out[a] = t[a];
#pragma unroll
        for (int a = 10; a < 16; ++a) out[a] = 0.0f;  // pad -> clean WMMA columns
    }
}

// Single-wave final reduction. EXEC is all-ones (32 threads, uniform control
// flow) as required by WMMA. Column-sum via D = ones(16x4) x B(4x16) + C:
// lane L supplies B slots in column n = L%16; the 4 k-slots of column n
// (lanes n and n+16, 2 VGPRs each) carry 4 disjoint row-groups of the
// partials, so D[*][n] = total for accumulator n, in f32 throughout.
__global__ void __launch_bounds__(32)
amp_final(const float* __restrict__ part, float* __restrict__ out,
          float batchN, int nblk) {
    const int L    = threadIdx.x;
    const int half = L >> 4;   // 0: k in {0,1}; 1: k in {2,3}
    const int n    = L & 15;   // column = accumulator index

    // per-lane pre-accumulation over block partials (rows), 2 B-slots per lane
    float bx = 0.0f, by = 0.0f;
    for (int b0 = 0; b0 < nblk; b0 += 4) {
        bx += part[(size_t)(b0 + 2 * half + 0) * 16 + n];
        by += part[(size_t)(b0 + 2 * half + 1) * 16 + n];
    }

    v2f a; a.x = 1.0f; a.y = 1.0f;         // A = ones(16x4)
    v2f b; b.x = bx;   b.y = by;           // B(4x16) residual partials
    v8f c = {};
    c = __builtin_amdgcn_wmma_f32_16x16x4_f32(
            /*neg_a=*/false, a, /*neg_b=*/false, b,
            /*c_mod=*/(short)0, c, /*reuse_a=*/false, /*reuse_b=*/false);

    __shared__ float tot[16];
    if (L < 16) tot[L] = c[0];             // every row of D == column sums
    __syncthreads();

    if (L == 0) {
        // tot[0..4] = counts, tot[5..9] = raw masked smape sums
        float rst = 0.0f, m_rst = 0.0f, mcnt = 0.0f;
#pragma unroll
        for (int i = 0; i < 5; ++i) {
            float cnt = tot[i];
            float w   = sqrtf(logf(batchN / fmaxf(cnt, 1.0f)));
            float S   = tot[5 + i] * w;
            if (cnt > 0.0f) {
                m_rst = (m_rst + S) / cnt * 0.5f;
                rst  += m_rst;
                mcnt += 1.0f;
            }
        }
        out[0] = (mcnt == 0.0f) ? (rst / 5.0f) : (rst / mcnt);
    }
}

extern "C" void kernel_launch(void* const* d_in, const int* in_sizes, int n_in,
                              void* d_out, int out_size, void* d_ws, size_t ws_size,
                              hipStream_t stream) {
    // setup_inputs order: dbp_pred, sbp_pred, mbp_pred(dead), d, s, m(dead)
    const float* dbp = (const float*)d_in[0];
    const float* sbp = (const float*)d_in[1];
    const float* dt  = (const float*)d_in[3];
    const float* st  = (const float*)d_in[4];
    const int B = in_sizes[0];

    // partials: nblk * 16 floats; nblk multiple of 4 for the 16x16x4 tile
    int nblk = NBLK;
    size_t need = (size_t)nblk * 16 * sizeof(float);
    while (nblk > 4 && need > ws_size) { nblk >>= 1; need = (size_t)nblk * 16 * sizeof(float); }
    float* part = (float*)d_ws;

    amp_partial<<<nblk, TPB, 0, stream>>>(dbp, sbp, dt, st, part, B / 4);
    amp_final<<<1, 32, 0, stream>>>(part, (float*)d_out, (float)B, nblk);
}